// HardPolarVoxelization_40235253629494
// MI455X (gfx1250) — compile-verified
//
#include <hip/hip_runtime.h>
#include <stdint.h>

// ---------------- problem constants (match reference) ----------------
#define N_RADIAL   48
#define N_ANGULAR  72
#define HBINS      (N_RADIAL * N_ANGULAR)   // 3456
#define MAX_PTS    64
#define MAX_VOX    3000
#define PT_C       5

#define SEG_PTS    2048        // points per wave-owned segment
#define CHUNK      256         // points staged to LDS per TDM transfer
#define INVALID_H  0xFFFFFFFFu

// d_out float layout (concatenated tuple, return order)
#define VOX_OFF    0
#define COORD_OFF  (MAX_VOX * MAX_PTS * PT_C)      // 960000
#define NUM_OFF    (COORD_OFF + MAX_VOX * 3)       // 969000
#define CEN_OFF    (NUM_OFF + MAX_VOX)             // 972000
#define OUT_FLOATS (CEN_OFF + MAX_VOX * 2)         // 978000

// float32 pi (0x40490FDB) and linspace step, constant-folded in f32 RNE
#define PI_F   3.14159265358979323846f
static constexpr float TWO_PI_F   = 2.0f * PI_F;          // exact (exp shift)
static constexpr float STEP_F     = TWO_PI_F / 72.0f;     // RNE at compile time
static constexpr float INV_STEP_F = 72.0f / TWO_PI_F;

// ---------------- helpers ----------------
__device__ __forceinline__ float edgef(int i) {
  // ANGLE_EDGES[i] = -pi + i*step, each op rounded in f32 (mimics linspace)
  return __fadd_rn(-PI_F, __fmul_rn((float)i, STEP_F));
}

// hash = r_idx * 72 + t_idx for valid points, INVALID_H otherwise.
// searchsorted(..., 'left')-1 semantics: equality on an edge goes to lower bin.
__device__ __forceinline__ unsigned point_hash(float x, float y, float z) {
  float r  = sqrtf(__fadd_rn(__fmul_rn(x, x), __fmul_rn(y, y)));
  float th = atan2f(y, x);
  float tEnd = edgef(N_ANGULAR);
  if (!((z >= -3.0f) & (z < 5.0f) & (r >= 2.0f) & (r < 50.0f) &
        (th >= -PI_F) & (th < tEnd)))
    return INVALID_H;
  // radial edges are exact ints 2..50: count(e < r) - 1 == ceil(r) - 3
  int ri = (int)ceilf(r) - 3;
  ri = ri < 0 ? 0 : (ri > N_RADIAL - 1 ? N_RADIAL - 1 : ri);
  // angular: estimate then correct against exact f32 edges
  int ti = (int)floorf(__fmul_rn(__fadd_rn(th, PI_F), INV_STEP_F));
  ti = ti < 0 ? 0 : (ti > N_ANGULAR - 1 ? N_ANGULAR - 1 : ti);
  while (ti > 0 && !(edgef(ti) < th)) --ti;                // equality -> lower bin
  while (ti < N_ANGULAR - 1 && edgef(ti + 1) < th) ++ti;
  return (unsigned)(ri * N_ANGULAR + ti);
}

// ---------------- TDM: issue nElem-float transfer (global -> LDS) ----------------
typedef unsigned int u32x4 __attribute__((ext_vector_type(4)));
typedef int          i32x8 __attribute__((ext_vector_type(8)));
typedef int          i32x4 __attribute__((ext_vector_type(4)));

#if __has_builtin(__builtin_amdgcn_tensor_load_to_lds)
#define HAVE_TDM 1
#else
#define HAVE_TDM 0
#endif

__device__ __forceinline__ void tdm_issue_f32(float* ldsDst, const float* gSrc, int nElem) {
#if HAVE_TDM
  // D# per CDNA5 ISA ch.8: group0 = {flags, lds_addr, gaddr_lo, gaddr_hi|type=2}
  uint32_t lds = (uint32_t)(uintptr_t)ldsDst;          // low 32 bits of LDS aperture addr
  uint64_t ga  = (uint64_t)(uintptr_t)gSrc;
  u32x4 g0 = { 1u,                                     // count=1, user mode
               lds,
               (uint32_t)ga,
               (uint32_t)((ga >> 32) & 0x01FFFFFFu) | (2u << 30) };  // type=2 (image)
  unsigned ne = (unsigned)nElem;
  i32x8 g1 = { (int)(2u << 16),                        // data_size=2 (4B), wg_mask=0
               (int)((ne & 0xFFFFu) << 16),            // tensor_dim0[15:0]
               (int)((ne >> 16) | (1u << 16)),         // tensor_dim0[31:16] | tensor_dim1=1
               (int)(ne << 16),                        // tile_dim0 = nElem (1-D tile)
               0,                                      // tile_dim1=0 (unused), tile_dim2=0
               (int)ne,                                // tensor_dim0_stride lo
               0, 0 };
  i32x4 z4 = { 0, 0, 0, 0 };
  i32x8 z8 = { 0, 0, 0, 0, 0, 0, 0, 0 };
  __builtin_amdgcn_tensor_load_to_lds(g0, g1, z4, z4, z8, 0);
#else
  for (int i = (int)(threadIdx.x & 31); i < nElem; i += 32) ldsDst[i] = gSrc[i];
#endif
}

// wait until at most Outstanding TDM ops remain (in-order per wave)
template <int Outstanding>
__device__ __forceinline__ void tdm_wait() {
#if HAVE_TDM && __has_builtin(__builtin_amdgcn_s_wait_tensorcnt)
  __builtin_amdgcn_s_wait_tensorcnt(Outstanding);
#endif
}

// ---------------- kernel 0: zero outputs ----------------
__global__ void zero_kernel(float* out, int n) {
  int i = blockIdx.x * blockDim.x + threadIdx.x;
  if (i < n) out[i] = 0.0f;
}

// ---------------- kernel 1: per-segment histogram (2 waves/block) ----------------
__global__ __launch_bounds__(64) void count_kernel(const float* __restrict__ pts,
                                                   int nPts, int numSeg,
                                                   uint32_t* __restrict__ segCounts) {
  __shared__ uint32_t cnt[2][HBINS];
  const int wave = threadIdx.x >> 5, lane = threadIdx.x & 31;
  const int seg = blockIdx.x * 2 + wave;
  uint32_t* c = cnt[wave];
  for (int h = lane; h < HBINS; h += 32) c[h] = 0u;
  __builtin_amdgcn_wave_barrier();
  if (seg >= numSeg) return;
  const int base = seg * SEG_PTS;
  const int nSeg = min(SEG_PTS, nPts - base);
  for (int i = lane; i < nSeg; i += 32) {
    const float* p = pts + (size_t)(base + i) * PT_C;
    __builtin_prefetch(p + 32 * PT_C, 0, 0);           // global_prefetch_b8
    unsigned h = point_hash(p[0], p[1], p[2]);
    if (h != INVALID_H) atomicAdd(&c[h], 1u);          // ds_add (order-free pass)
  }
  __builtin_amdgcn_wave_barrier();
  for (int h = lane; h < HBINS; h += 32)
    segCounts[(size_t)seg * HBINS + h] = c[h];         // bin-minor: coalesced
}

// ---------------- kernel 2: per-hash exclusive scan over segments ----------------
__global__ void scan_kernel(uint32_t* __restrict__ segCounts, int numSeg,
                            uint32_t* __restrict__ totals) {
  int h = blockIdx.x * blockDim.x + threadIdx.x;
  if (h >= HBINS) return;
  uint32_t run = 0;
  for (int s = 0; s < numSeg; s++) {                   // coalesced across threads
    size_t idx = (size_t)s * HBINS + h;
    uint32_t v = segCounts[idx];
    segCounts[idx] = run;                              // becomes exclusive base
    run += v;
  }
  totals[h] = run;
}

// ---------------- kernel 3: slot compaction + coords/num/centers ----------------
#define SLOT_ITEMS ((HBINS + 255) / 256)               // 14 contiguous bins/thread
__global__ __launch_bounds__(256) void slots_kernel(const uint32_t* __restrict__ totals,
                                                    int* __restrict__ hash2slot,
                                                    float* __restrict__ out) {
  __shared__ int psum[256];
  const int t = threadIdx.x;
  const int h0 = t * SLOT_ITEMS;
  int cnt = 0;
  for (int j = 0; j < SLOT_ITEMS; j++) {
    int h = h0 + j;
    if (h < HBINS && totals[h] > 0u) cnt++;
  }
  psum[t] = cnt;
  __syncthreads();
  for (int off = 1; off < 256; off <<= 1) {            // inclusive Hillis-Steele
    int add = (t >= off) ? psum[t - off] : 0;
    __syncthreads();
    psum[t] += add;
    __syncthreads();
  }
  int slot = psum[t] - cnt;                            // exclusive base
  for (int j = 0; j < SLOT_ITEMS; j++) {
    int h = h0 + j;
    if (h >= HBINS) break;
    uint32_t tot = totals[h];
    if (tot > 0u) {
      if (slot < MAX_VOX) {
        hash2slot[h] = slot;
        int ri = h / N_ANGULAR, ti = h % N_ANGULAR;
        out[COORD_OFF + slot * 3 + 0] = (float)ri;
        out[COORD_OFF + slot * 3 + 1] = (float)ti;
        out[COORD_OFF + slot * 3 + 2] = 0.0f;
        out[NUM_OFF + slot] = (float)min((int)tot, MAX_PTS);
        float r_c = (float)ri + 2.5f;                  // 0.5*((2+ri)+(3+ri))
        float t_c = __fmul_rn(0.5f, __fadd_rn(edgef(ti), edgef(ti + 1)));
        out[CEN_OFF + slot * 2 + 0] = __fmul_rn(r_c, cosf(t_c));
        out[CEN_OFF + slot * 2 + 1] = __fmul_rn(r_c, sinf(t_c));
      } else {
        hash2slot[h] = 0x7FFFFFFF;
      }
      slot++;
    } else {
      hash2slot[h] = 0x7FFFFFFF;
    }
  }
}

// ------- kernel 4: ordered scatter (double-buffered TDM pipeline, 2 waves/blk) ----
__global__ __launch_bounds__(64) void scatter_kernel(const float* __restrict__ pts,
                                                     int nPts, int numSeg,
                                                     const uint32_t* __restrict__ segBase,
                                                     const int* __restrict__ hash2slot,
                                                     float* __restrict__ out) {
  __shared__ uint32_t cnt[2][HBINS];
  __shared__ float    stage[2][2][CHUNK * PT_C];       // [wave][buffer]
  const int wave = threadIdx.x >> 5, lane = threadIdx.x & 31;
  const int seg = blockIdx.x * 2 + wave;
  if (seg >= numSeg) return;
  uint32_t* c = cnt[wave];
  for (int h = lane; h < HBINS; h += 32)
    c[h] = segBase[(size_t)seg * HBINS + h];           // global exclusive bases
  __builtin_amdgcn_wave_barrier();
  const int base = seg * SEG_PTS;
  const int nSeg = min(SEG_PTS, nPts - base);
  const int nChunks = (nSeg + CHUNK - 1) / CHUNK;

  // pipeline prologue: start DMA of chunk 0
  tdm_issue_f32(stage[wave][0], pts + (size_t)base * PT_C, min(CHUNK, nSeg) * PT_C);

  for (int ck = 0; ck < nChunks; ck++) {
    const int cs = ck * CHUNK;
    const int n  = min(CHUNK, nSeg - cs);
    if (ck + 1 < nChunks) {
      // issue next chunk into the other buffer, then wait for the oldest only
      const int cs1 = cs + CHUNK;
      tdm_issue_f32(stage[wave][(ck + 1) & 1],
                    pts + (size_t)(base + cs1) * PT_C,
                    min(CHUNK, nSeg - cs1) * PT_C);
      tdm_wait<1>();                                   // s_wait_tensorcnt 1: chunk ck ready
    } else {
      tdm_wait<0>();                                   // last chunk: drain
    }
    __builtin_amdgcn_wave_barrier();
    const float* st = stage[wave][ck & 1];
    for (int i = 0; i < CHUNK; i += 32) {
      const int li = i + lane;
      unsigned h = INVALID_H;
      float p0 = 0.f, p1 = 0.f, p2 = 0.f, p3 = 0.f, p4 = 0.f;
      if (li < n) {
        p0 = st[li * PT_C + 0]; p1 = st[li * PT_C + 1]; p2 = st[li * PT_C + 2];
        p3 = st[li * PT_C + 3]; p4 = st[li * PT_C + 4];
        h = point_hash(p0, p1, p2);
      }
      // stable within-chunk rank: lanes < me with equal hash (wave32 match loop)
      unsigned cprev = 0; bool isLast = true;
      #pragma unroll
      for (int k = 0; k < 32; k++) {
        unsigned hk = (unsigned)__shfl((int)h, k, 32);
        if (hk == h) { if (k < lane) cprev++; else if (k > lane) isLast = false; }
      }
      if (h != INVALID_H) {
        uint32_t b = c[h];                             // all lanes read first (lockstep)
        uint32_t rank = b + cprev;
        if (isLast) c[h] = rank + 1u;                  // one writer per hash value
        if (rank < MAX_PTS) {
          int slot = hash2slot[h];
          if (slot < MAX_VOX) {
            float* dst = out + VOX_OFF + ((size_t)slot * MAX_PTS + rank) * PT_C;
            dst[0] = p0; dst[1] = p1; dst[2] = p2; dst[3] = p3; dst[4] = p4;
          }
        }
      }
      __builtin_amdgcn_wave_barrier();                 // order LDS counter RMW phases
    }
  }
}

// ---------------- launch ----------------
extern "C" void kernel_launch(void* const* d_in, const int* in_sizes, int n_in,
                              void* d_out, int out_size, void* d_ws, size_t ws_size,
                              hipStream_t stream) {
  const float* pts = (const float*)d_in[0];
  const int nPts = in_sizes[0] / PT_C;
  const int numSeg = (nPts + SEG_PTS - 1) / SEG_PTS;

  uint32_t* segCounts = (uint32_t*)d_ws;                       // numSeg*HBINS u32
  uint32_t* totals    = segCounts + (size_t)numSeg * HBINS;    // HBINS u32
  int*      hash2slot = (int*)(totals + HBINS);                // HBINS i32
  float*    out       = (float*)d_out;

  zero_kernel<<<(out_size + 255) / 256, 256, 0, stream>>>(out, out_size);
  count_kernel<<<(numSeg + 1) / 2, 64, 0, stream>>>(pts, nPts, numSeg, segCounts);
  scan_kernel<<<(HBINS + 255) / 256, 256, 0, stream>>>(segCounts, numSeg, totals);
  slots_kernel<<<1, 256, 0, stream>>>(totals, hash2slot, out);
  scatter_kernel<<<(numSeg + 1) / 2, 64, 0, stream>>>(pts, nPts, numSeg,
                                                      segCounts, hash2slot, out);
}